// _ScaledDotProductAttention_15083925143887
// MI455X (gfx1250) — compile-verified
//
#include <hip/hip_runtime.h>

typedef __attribute__((ext_vector_type(2))) float v2f;
typedef __attribute__((ext_vector_type(8))) float v8f;

#define B_ 64
#define H_ 16
#define L_ 512
#define D_ 32
#define MT 16                     // rows of S/out per block
#define NWAVES 4
#define COLS_PER_WAVE (L_ / NWAVES)  // 128
#define NTILES (COLS_PER_WAVE / 16)  // 8
#define PSTR 516                  // padded LDS row stride (floats) -> conflict-free frag loads
#define OSTR 33                   // padded O row stride

__global__ __launch_bounds__(128) void attn_f32wmma_kernel(
    const float* __restrict__ q,      // [B,H,L,D]
    const float* __restrict__ k,      // [B,H,D,L]  (already B-oriented for QK^T)
    const float* __restrict__ v,      // [B,H,L,D]
    const float* __restrict__ mat,    // [B,H,L,L]
    const float* __restrict__ scale,  // [1]
    float* __restrict__ out,          // [B,H,L,D]
    float* __restrict__ scores)       // [B,H,L,L]
{
    __shared__ float lds_p[MT * PSTR];       // normalized P, padded rows
    __shared__ float lds_red[NWAVES][MT];    // cross-wave row max / sum
    __shared__ float lds_o[MT * OSTR];       // O accumulator

    const int tid  = threadIdx.x;
    const int lane = tid & 31;
    const int wv   = tid >> 5;
    const int hlf  = lane >> 4;   // which 16-lane half
    const int l16  = lane & 15;

    const int mtiles = L_ / MT;                  // 32
    const int bh = blockIdx.x / mtiles;
    const int mt = blockIdx.x % mtiles;
    const int m0 = mt * MT;

    const float* qb = q   + (size_t)bh * L_ * D_;
    const float* kb = k   + (size_t)bh * D_ * L_;
    const float* vb = v   + (size_t)bh * L_ * D_;
    const float* mb = mat + (size_t)bh * L_ * L_;
    float* scb = scores + (size_t)bh * L_ * L_;
    float* ob  = out    + (size_t)bh * L_ * D_;

    const float sc = scale[0];

    // zero O accumulator
    for (int i = tid; i < MT * OSTR; i += 128) lds_o[i] = 0.0f;

    // ---------------- Phase 1: S = (Q K^T) * matrix * scale ----------------
    // A-fragments (Q rows m0..m0+15, all of D): lane<16 holds (row=l16, K=d0,d0+1),
    // lane>=16 holds (row=l16, K=d0+2,d0+3)  -> single b64 load each.
    v2f afr[8];
    {
        const float* qrow = qb + (size_t)(m0 + l16) * D_ + 2 * hlf;
        #pragma unroll
        for (int d = 0; d < 8; ++d)
            afr[d] = *(const v2f*)(qrow + 4 * d);
    }

    v8f  sreg[NTILES];
    float rmax[8];
    #pragma unroll
    for (int j = 0; j < 8; ++j) rmax[j] = -3.4e38f;

    const int col_base = wv * COLS_PER_WAVE;
    #pragma unroll
    for (int t = 0; t < NTILES; ++t) {
        const int c0 = col_base + t * 16;
        v8f acc = {};
        #pragma unroll
        for (int d = 0; d < 8; ++d) {
            const int d0 = 4 * d;
            // B-fragment: lane holds column N=c0+l16, K rows d0+2*hlf, d0+2*hlf+1
            v2f bfr;
            bfr.x = kb[(size_t)(d0 + 2 * hlf)     * L_ + c0 + l16];
            bfr.y = kb[(size_t)(d0 + 2 * hlf + 1) * L_ + c0 + l16];
            acc = __builtin_amdgcn_wmma_f32_16x16x4_f32(false, afr[d], false, bfr,
                                                        (short)0, acc, false, false);
        }
        // elementwise matrix * scale, write scores, track row max
        #pragma unroll
        for (int j = 0; j < 8; ++j) {
            const int row = m0 + j + 8 * hlf;            // C layout: M = j + 8*half
            const size_t idx = (size_t)row * L_ + c0 + l16;
            float sv = acc[j] * mb[idx] * sc;
            scb[idx] = sv;
            acc[j] = sv;
            rmax[j] = fmaxf(rmax[j], sv);
        }
        sreg[t] = acc;
    }

    // ---------------- Phase 2: softmax over 512 columns ----------------
    // row max: reduce across the 16 lanes holding the same row
    #pragma unroll
    for (int j = 0; j < 8; ++j) {
        float m = rmax[j];
        #pragma unroll
        for (int off = 1; off < 16; off <<= 1)
            m = fmaxf(m, __shfl_xor(m, off, 32));
        rmax[j] = m;
    }
    if (l16 == 0) {
        #pragma unroll
        for (int j = 0; j < 8; ++j) lds_red[wv][8 * hlf + j] = rmax[j];
    }
    __syncthreads();
    float rowM[8];
    #pragma unroll
    for (int j = 0; j < 8; ++j) {
        const int r = 8 * hlf + j;
        rowM[j] = fmaxf(fmaxf(lds_red[0][r], lds_red[1][r]),
                        fmaxf(lds_red[2][r], lds_red[3][r]));
    }
    __syncthreads();

    // exp + row sum
    float rsum[8];
    #pragma unroll
    for (int j = 0; j < 8; ++j) rsum[j] = 0.0f;
    #pragma unroll
    for (int t = 0; t < NTILES; ++t) {
        #pragma unroll
        for (int j = 0; j < 8; ++j) {
            float e = __expf(sreg[t][j] - rowM[j]);
            sreg[t][j] = e;
            rsum[j] += e;
        }
    }
    #pragma unroll
    for (int j = 0; j < 8; ++j) {
        float s = rsum[j];
        #pragma unroll
        for (int off = 1; off < 16; off <<= 1)
            s += __shfl_xor(s, off, 32);
        rsum[j] = s;
    }
    if (l16 == 0) {
        #pragma unroll
        for (int j = 0; j < 8; ++j) lds_red[wv][8 * hlf + j] = rsum[j];
    }
    __syncthreads();
    float rinv[8];
    #pragma unroll
    for (int j = 0; j < 8; ++j) {
        const int r = 8 * hlf + j;
        rinv[j] = 1.0f / (lds_red[0][r] + lds_red[1][r] + lds_red[2][r] + lds_red[3][r]);
    }

    // normalized P -> LDS (row-major, padded stride)
    #pragma unroll
    for (int t = 0; t < NTILES; ++t) {
        const int c0 = col_base + t * 16;
        #pragma unroll
        for (int j = 0; j < 8; ++j) {
            const int r = 8 * hlf + j;
            lds_p[r * PSTR + c0 + l16] = sreg[t][j] * rinv[j];
        }
    }
    __syncthreads();

    // ---------------- Phase 3: O = P @ V via WMMA ----------------
    v8f oacc0 = {};
    v8f oacc1 = {};
    #pragma unroll 4
    for (int s = 0; s < COLS_PER_WAVE / 4; ++s) {
        const int k0 = col_base + 4 * s;
        // A-fragment of P from LDS: lane holds (row=l16, K=k0+2*hlf, +1) -> b64, conflict-free
        v2f pa = *(const v2f*)&lds_p[l16 * PSTR + k0 + 2 * hlf];
        // B-fragments of V (row-major [L,D]): column N = nt*16+l16, K rows k0+2*hlf, +1
        v2f bv0, bv1;
        bv0.x = vb[(size_t)(k0 + 2 * hlf)     * D_ + l16];
        bv0.y = vb[(size_t)(k0 + 2 * hlf + 1) * D_ + l16];
        bv1.x = vb[(size_t)(k0 + 2 * hlf)     * D_ + 16 + l16];
        bv1.y = vb[(size_t)(k0 + 2 * hlf + 1) * D_ + 16 + l16];
        oacc0 = __builtin_amdgcn_wmma_f32_16x16x4_f32(false, pa, false, bv0,
                                                      (short)0, oacc0, false, false);
        oacc1 = __builtin_amdgcn_wmma_f32_16x16x4_f32(false, pa, false, bv1,
                                                      (short)0, oacc1, false, false);
    }

    // reduce partial O across the 4 waves with LDS float atomics
    #pragma unroll
    for (int j = 0; j < 8; ++j) {
        const int r = 8 * hlf + j;
        atomicAdd(&lds_o[r * OSTR + l16],      oacc0[j]);
        atomicAdd(&lds_o[r * OSTR + 16 + l16], oacc1[j]);
    }
    __syncthreads();

    // write O
    for (int i = tid; i < MT * D_; i += 128) {
        const int r = i / D_, c = i % D_;
        ob[(size_t)(m0 + r) * D_ + c] = lds_o[r * OSTR + c];
    }
}

extern "C" void kernel_launch(void* const* d_in, const int* in_sizes, int n_in,
                              void* d_out, int out_size, void* d_ws, size_t ws_size,
                              hipStream_t stream) {
    (void)in_sizes; (void)n_in; (void)out_size; (void)d_ws; (void)ws_size;
    const float* q     = (const float*)d_in[0];
    const float* k     = (const float*)d_in[1];
    const float* v     = (const float*)d_in[2];
    const float* mat   = (const float*)d_in[3];
    const float* scale = (const float*)d_in[4];

    float* out    = (float*)d_out;                              // [B,H,L,D]
    float* scores = out + (size_t)B_ * H_ * L_ * D_;            // [B,H,L,L]

    dim3 grid(B_ * H_ * (L_ / MT));   // 32768 blocks
    dim3 block(128);                  // 4 waves
    attn_f32wmma_kernel<<<grid, block, 0, stream>>>(q, k, v, mat, scale, out, scores);
}